// NCESoftmaxLoss_var_1477468750344
// MI455X (gfx1250) — compile-verified
//
#include <hip/hip_runtime.h>
#include <hip/hip_bf16.h>

// ---------------------------------------------------------------------------
// Problem constants (from reference)
// ---------------------------------------------------------------------------
#define NUM_CLASSES 4096
#define NUM_GROUPS  128
#define ROW_THREADS 256                       // 8 waves of 32 (wave32 on gfx1250)
#define PER_THREAD  (NUM_CLASSES / ROW_THREADS) // 16 f32 per thread, kept in regs
#define WAVES_PER_BLOCK (ROW_THREADS / 32)

typedef __attribute__((ext_vector_type(2))) float v2f;
typedef __attribute__((ext_vector_type(8))) float v8f;

// ---------------------------------------------------------------------------
// Wave-level reductions (wave32)
// ---------------------------------------------------------------------------
__device__ __forceinline__ float wave_max32(float v) {
#pragma unroll
  for (int off = 16; off > 0; off >>= 1)
    v = fmaxf(v, __shfl_xor(v, off, 32));
  return v;
}

// Sum 32 fp32 lane partials with V_WMMA_F32_16X16X4_F32.
// A (16x4 f32, 2 VGPRs): VGPR0 = partial -> A[M][0] (lanes 0-15), A[M][2]
// (lanes 16-31); VGPR1 = 0. B = all ones (layout-independent). C = 0.
// => D[i][j] = partial[i] + partial[i+16].
// Lane j holds D[0..7][j] in c[0..7]; lane j+16 holds D[8..15][j].
// Per-lane sum of c[] + one SWAPX16 shuffle -> full 32-lane sum in all lanes.
__device__ __forceinline__ float wave_sum32(float p) {
#if __has_builtin(__builtin_amdgcn_wmma_f32_16x16x4_f32)
  v2f a; a.x = p;   a.y = 0.0f;
  v2f b; b.x = 1.0f; b.y = 1.0f;
  v8f c = {};
  c = __builtin_amdgcn_wmma_f32_16x16x4_f32(
      /*neg_a=*/false, a, /*neg_b=*/false, b,
      /*c_mod=*/(short)0, c, /*reuse_a=*/false, /*reuse_b=*/false);
  float s = ((c[0] + c[1]) + (c[2] + c[3])) + ((c[4] + c[5]) + (c[6] + c[7]));
  s += __shfl_xor(s, 16, 32);   // combine column halves: total in every lane
  return s;
#else
#pragma unroll
  for (int off = 16; off > 0; off >>= 1) p += __shfl_xor(p, off, 32);
  return p;
#endif
}

// ---------------------------------------------------------------------------
// Kernel 1: zero the presence bitmap (workspace is poisoned, never re-zeroed)
// ---------------------------------------------------------------------------
__global__ void zero_present_kernel(int* __restrict__ present) {
  present[threadIdx.x] = 0;
}

// ---------------------------------------------------------------------------
// Kernel 2: mark which group ids occur in graph_idx (benign write races)
// ---------------------------------------------------------------------------
__global__ void mark_present_kernel(const int* __restrict__ gidx, int n,
                                    int* __restrict__ present) {
  int i = blockIdx.x * blockDim.x + threadIdx.x;
  if (i < n) {
    int g = gidx[i];
    if ((unsigned)g < (unsigned)NUM_GROUPS) present[g] = 1;
  }
}

// ---------------------------------------------------------------------------
// Kernel 3: loss[g] = logsumexp(x[g,:]) - x[g,0] for all 128 candidate rows.
// One block per row; 4096 f32 per row held entirely in registers (16/thread).
// Two-pass: block max, then sum of exp(v - max). WMMA does the wave sums.
// EXEC is all-ones at the WMMA (uniform control flow, blockDim % 32 == 0).
// ---------------------------------------------------------------------------
__global__ void row_logsumexp_kernel(const float* __restrict__ x,
                                     float* __restrict__ loss) {
  const int g = blockIdx.x;
  const float* __restrict__ row = x + (size_t)g * NUM_CLASSES;
  const int t = threadIdx.x;

  __builtin_prefetch(row + t, 0, 0);  // global_prefetch_b8: pull row into cache

  float v[PER_THREAD];
#pragma unroll
  for (int i = 0; i < PER_THREAD; ++i)      // fully coalesced, stride 1 KB
    v[i] = row[t + i * ROW_THREADS];

  // ---- pass 1: block max ----
  float m = v[0];
#pragma unroll
  for (int i = 1; i < PER_THREAD; ++i) m = fmaxf(m, v[i]);
  m = wave_max32(m);

  __shared__ float smax[WAVES_PER_BLOCK];
  __shared__ float ssum[WAVES_PER_BLOCK];
  const int wave = t >> 5;
  const int lane = t & 31;
  if (lane == 0) smax[wave] = m;
  __syncthreads();

  float bmax = smax[0];
#pragma unroll
  for (int w = 1; w < WAVES_PER_BLOCK; ++w) bmax = fmaxf(bmax, smax[w]);

  // ---- pass 2: sum of exp(v - max), wave sum via v_wmma_f32_16x16x4_f32 ----
  float s = 0.0f;
#pragma unroll
  for (int i = 0; i < PER_THREAD; ++i) s += expf(v[i] - bmax);
  s = wave_sum32(s);

  if (lane == 0) ssum[wave] = s;
  __syncthreads();

  if (t == 0) {
    float tot = 0.0f;
#pragma unroll
    for (int w = 0; w < WAVES_PER_BLOCK; ++w) tot += ssum[w];
    // loss = -log_softmax(x)[0] = logsumexp(x) - x[0]
    loss[g] = bmax + logf(tot) - row[0];
  }
}

// ---------------------------------------------------------------------------
// Kernel 4: unbiased variance over losses of present groups + flag.
// Single block of 128 threads. Select (not multiply) masks out stale values.
// ---------------------------------------------------------------------------
__global__ void finalize_kernel(const int* __restrict__ present,
                                const float* __restrict__ loss,
                                float* __restrict__ out) {
  __shared__ float redf[NUM_GROUPS];
  __shared__ int   redi[NUM_GROUPS];
  __shared__ float s_mean;
  __shared__ int   s_count;

  const int t = threadIdx.x;
  const int p = present[t];
  const float L = p ? loss[t] : 0.0f;

  redf[t] = L;
  redi[t] = p;
  __syncthreads();
  for (int sft = NUM_GROUPS / 2; sft > 0; sft >>= 1) {
    if (t < sft) { redf[t] += redf[t + sft]; redi[t] += redi[t + sft]; }
    __syncthreads();
  }
  if (t == 0) {
    s_count = redi[0];
    s_mean  = redf[0] / (float)redi[0];
  }
  __syncthreads();

  const float mean = s_mean;
  const int count  = s_count;
  const float d = L - mean;
  redf[t] = p ? d * d : 0.0f;
  __syncthreads();
  for (int sft = NUM_GROUPS / 2; sft > 0; sft >>= 1) {
    if (t < sft) redf[t] += redf[t + sft];
    __syncthreads();
  }
  if (t == 0) {
    out[0] = redf[0] / (float)(count - 1);  // ddof=1 (NaN if count==1, as jnp)
    out[1] = (count == 1) ? 1.0f : 0.0f;
  }
}

// ---------------------------------------------------------------------------
// Launch
// ---------------------------------------------------------------------------
extern "C" void kernel_launch(void* const* d_in, const int* in_sizes, int n_in,
                              void* d_out, int out_size, void* d_ws, size_t ws_size,
                              hipStream_t stream) {
  const float* x    = (const float*)d_in[0];  // [16384, 4096] f32
  const int*   gidx = (const int*)d_in[1];    // [16384] group ids in [0,128)
  const int n = in_sizes[1];

  int*   present = (int*)d_ws;
  float* loss    = (float*)((char*)d_ws + NUM_GROUPS * sizeof(int));
  float* out     = (float*)d_out;

  zero_present_kernel<<<1, NUM_GROUPS, 0, stream>>>(present);
  mark_present_kernel<<<(n + 255) / 256, 256, 0, stream>>>(gidx, n, present);
  row_logsumexp_kernel<<<NUM_GROUPS, ROW_THREADS, 0, stream>>>(x, loss);
  finalize_kernel<<<1, NUM_GROUPS, 0, stream>>>(present, loss, out);
}